// SummGCN_25091198943314
// MI455X (gfx1250) — compile-verified
//
#include <hip/hip_runtime.h>
#include <hip/hip_bf16.h>

// ---------------------------------------------------------------------------
// 2-layer GCN forward on gfx1250 (wave32, WMMA f32 16x16x4, async LDS staging)
//   Z1 = x @ W1                       [50000 x 256]
//   h  = spmm(Z1) + b1   (relu fused into GEMM2's A-fragment read)
//   Z2 = relu(h) @ W2                 [50000 x 64]
//   out= spmm(Z2) + b2
// ---------------------------------------------------------------------------

typedef float v2f __attribute__((ext_vector_type(2)));
typedef float v8f __attribute__((ext_vector_type(8)));
typedef int   v4i __attribute__((vector_size(16)));   // matches builtin proto

#define KC  64          // K chunk staged in LDS
#define LDA 68          // A LDS row stride (floats): 16B-aligned rows, bank = 4m+k

#ifdef __has_builtin
#if __has_builtin(__builtin_amdgcn_global_load_async_to_lds_b128) && \
    __has_builtin(__builtin_amdgcn_s_wait_asynccnt)
#define HAVE_ASYNC_LDS 1
#endif
#endif
#ifndef HAVE_ASYNC_LDS
#define HAVE_ASYNC_LDS 0
#endif

// 16B global -> LDS copy: async (ASYNCcnt, no VGPR staging) when available.
__device__ __forceinline__ void copy16_g2l(const float* gsrc, float* ldst) {
#if HAVE_ASYNC_LDS
    __builtin_amdgcn_global_load_async_to_lds_b128(
        (__attribute__((address_space(1))) v4i*)gsrc,
        (__attribute__((address_space(3))) v4i*)ldst, 0, 0);
#else
    *(float4*)ldst = *(const float4*)gsrc;
#endif
}

__device__ __forceinline__ void wait_async_copies() {
#if HAVE_ASYNC_LDS
    __builtin_amdgcn_s_wait_asynccnt(0);
#endif
}

// ---------------------------------------------------------------------------
// GEMM: C[M x N] = op(A)[M x K] @ B[K x N],  op = relu if RELU_A.
// Block: NW waves, tile = 32 rows x 16*NW cols. Per wave: one 16-col tile,
// two 16-row subtiles. Double-buffered async LDS staging of A and B chunks.
// WMMA f32 16x16x4 fragment maps (ISA 7.12.2):
//   A (lane l): m = l&15, kpair = (l>>4)*2 -> {A[m][k], A[m][k+1]}
//   B (lane l): n = l&15, kpair = (l>>4)*2 -> {B[k][n], B[k+1][n]}
//   C/D (lane l): n = l&15, rows m = (l>>4)*8 + r, r = 0..7
// ---------------------------------------------------------------------------
template <int NW, bool RELU_A>
__global__ __launch_bounds__(NW * 32) void gemm_wmma(const float* __restrict__ A,
                                                     const float* __restrict__ B,
                                                     float* __restrict__ C,
                                                     int M, int K, int N) {
    constexpr int NB  = 16 * NW;      // cols per block
    constexpr int NBP = NB + 4;       // padded B LDS stride (mult of 4)

    __shared__ __align__(16) float lA[2][32 * LDA];
    __shared__ __align__(16) float lB[2][KC * NBP];

    const int tid   = threadIdx.x;
    const int lane  = tid & 31;
    const int wave  = tid >> 5;
    const int mBase = blockIdx.x * 32;
    const int nBase = blockIdx.y * NB;

    const int ml    = lane & 15;      // M (A frag) / N (B,C frags) index
    const int khalf = lane >> 4;      // selects K pair
    const int wn    = wave * 16 + ml; // lane's column within block tile

    // ---- stagers: cooperative 16B granule copies into LDS buffer `buf` ----
    auto stageA = [&](int buf, int kc) {
        constexpr int ITEMS = 32 * (KC / 4);
        for (int idx = tid; idx < ITEMS; idx += NW * 32) {
            const int r = idx >> 4;               // 0..31
            const int c = (idx & 15) << 2;        // 0..60 step 4
            int row = mBase + r;
            row = row < M ? row : M - 1;          // clamp tail (rows never stored)
            copy16_g2l(A + (size_t)row * K + kc + c, &lA[buf][r * LDA + c]);
        }
    };
    auto stageB = [&](int buf, int kc) {
        constexpr int ITEMS = KC * (NB / 4);
        for (int idx = tid; idx < ITEMS; idx += NW * 32) {
            const int kb = idx / (NB / 4);        // 0..KC-1
            const int c  = (idx % (NB / 4)) << 2; // 0..NB-4 step 4
            copy16_g2l(B + (size_t)(kc + kb) * N + nBase + c,
                       &lB[buf][kb * NBP + c]);
        }
    };

    v8f acc0 = {};
    v8f acc1 = {};

    stageA(0, 0);
    stageB(0, 0);
    wait_async_copies();
    __syncthreads();

    int cur = 0;
    for (int kc = 0; kc < K; kc += KC) {
        // kick off next chunk while we compute on this one
        if (kc + KC < K) {
            stageA(cur ^ 1, kc + KC);
            stageB(cur ^ 1, kc + KC);
        }

        const float* __restrict__ a0p = &lA[cur][ml * LDA];
        const float* __restrict__ a1p = &lA[cur][(ml + 16) * LDA];
        const float* __restrict__ bp  = &lB[cur][wn];

#pragma unroll
        for (int kk = 0; kk < KC; kk += 4) {
            const int kb = kk + khalf * 2;
            v2f a0 = *(const v2f*)(a0p + kb);
            v2f a1 = *(const v2f*)(a1p + kb);
            if (RELU_A) {
                a0.x = fmaxf(a0.x, 0.f); a0.y = fmaxf(a0.y, 0.f);
                a1.x = fmaxf(a1.x, 0.f); a1.y = fmaxf(a1.y, 0.f);
            }
            v2f b;
            b.x = bp[(size_t)kb * NBP];
            b.y = bp[(size_t)(kb + 1) * NBP];
            acc0 = __builtin_amdgcn_wmma_f32_16x16x4_f32(
                false, a0, false, b, (short)0, acc0, false, false);
            acc1 = __builtin_amdgcn_wmma_f32_16x16x4_f32(
                false, a1, false, b, (short)0, acc1, false, false);
        }

        wait_async_copies();   // our async writes to the other buffer are done
        __syncthreads();       // everyone finished reading `cur` / writing next
        cur ^= 1;
    }

    // ---- store D tiles ----
    const int col = nBase + wave * 16 + ml;
    const int mr  = khalf * 8;
#pragma unroll
    for (int r = 0; r < 8; ++r) {
        int row = mBase + mr + r;
        if (row < M) C[(size_t)row * N + col] = acc0[r];
        row += 16;
        if (row < M) C[(size_t)row * N + col] = acc1[r];
    }
}

// ---------------------------------------------------------------------------
// out[i*D + j] = bias[j]   (accumulator init; ws/out are poisoned by harness)
// ---------------------------------------------------------------------------
__global__ void init_bias(float* __restrict__ out, const float* __restrict__ b,
                          int nRows, int D) {
    const size_t total = (size_t)nRows * D;
    for (size_t i = (size_t)blockIdx.x * blockDim.x + threadIdx.x; i < total;
         i += (size_t)gridDim.x * blockDim.x) {
        out[i] = b[i % D];
    }
}

// ---------------------------------------------------------------------------
// Edge-parallel SpMM accumulate: out[rows[e]] += vals[e] * Z[cols[e]]
// D/4 lanes per edge, float4 gather, 4x global_atomic_add_f32 scatter.
// Gather source and scatter target are L2-resident (<< 192MB).
// ---------------------------------------------------------------------------
template <int D>
__global__ __launch_bounds__(256) void spmm_accum(const int* __restrict__ rows,
                                                  const int* __restrict__ cols,
                                                  const float* __restrict__ vals,
                                                  const float* __restrict__ Z,
                                                  float* __restrict__ out,
                                                  int E) {
    constexpr int TPE = D / 4;            // threads per edge
    constexpr int EPB = 256 / TPE;        // edges per block
    const int tid = threadIdx.x;
    const int e   = blockIdx.x * EPB + tid / TPE;
    if (e >= E) return;

    const int   r = rows[e];
    const int   c = cols[e];
    const float v = vals[e];
    const int   f = (tid % TPE) * 4;

    const float4 z = *(const float4*)(Z + (size_t)c * D + f);
    float* o = out + (size_t)r * D + f;
    atomicAdd(o + 0, v * z.x);
    atomicAdd(o + 1, v * z.y);
    atomicAdd(o + 2, v * z.z);
    atomicAdd(o + 3, v * z.w);
}

// ---------------------------------------------------------------------------
extern "C" void kernel_launch(void* const* d_in, const int* in_sizes, int n_in,
                              void* d_out, int out_size, void* d_ws, size_t ws_size,
                              hipStream_t stream) {
    const float* x    = (const float*)d_in[0];
    const int*   rows = (const int*)d_in[1];
    const int*   cols = (const int*)d_in[2];
    const float* vals = (const float*)d_in[3];
    const float* W1   = (const float*)d_in[4];
    const float* b1   = (const float*)d_in[5];
    const float* W2   = (const float*)d_in[6];
    const float* b2   = (const float*)d_in[7];
    float*       out  = (float*)d_out;

    const int N  = 50000;
    const int E  = in_sizes[1];           // 800000
    const int IN = 512, HID = 256, OUT = 64;

    float* Z1 = (float*)d_ws;                       // N*HID
    float* h  = Z1 + (size_t)N * HID;               // N*HID
    float* Z2 = h + (size_t)N * HID;                // N*OUT

    const int mTiles = (N + 31) / 32;

    // GEMM1: Z1 = x @ W1   (8 waves -> 128 cols/block, grid.y = 2)
    gemm_wmma<8, false><<<dim3(mTiles, HID / 128), 256, 0, stream>>>(
        x, W1, Z1, N, IN, HID);

    // h = b1 (broadcast), then += spmm(Z1)
    init_bias<<<2048, 256, 0, stream>>>(h, b1, N, HID);
    spmm_accum<HID><<<(E + 3) / 4, 256, 0, stream>>>(rows, cols, vals, Z1, h, E);

    // GEMM2: Z2 = relu(h) @ W2  (4 waves -> 64 cols/block, grid.y = 1)
    gemm_wmma<4, true><<<dim3(mTiles, 1), 128, 0, stream>>>(h, W2, Z2, N, HID, OUT);

    // out = b2 (broadcast), then += spmm(Z2)
    init_bias<<<1024, 256, 0, stream>>>(out, b2, N, OUT);
    spmm_accum<OUT><<<(E + 15) / 16, 256, 0, stream>>>(rows, cols, vals, Z2, out, E);
}